// Model_22960895164747
// MI455X (gfx1250) — compile-verified
//
#include <hip/hip_runtime.h>
#include <math.h>

#define GAMMA 0.1f
#define TAU   0.01f

typedef __attribute__((ext_vector_type(2))) float v2f;
typedef __attribute__((ext_vector_type(8))) float v8f;

__device__ __forceinline__ float sigmoidf(float v) {
    return 1.0f / (1.0f + expf(-v));
}

// Kernel 1: x_new = 0.9*x + 0.1*(W@phi(x)) + W_in_1@I ; phi_xn ; psi_z = tanh(z)
__global__ void k1_state(const float* __restrict__ I, const float* __restrict__ x,
                         const float* __restrict__ W, const float* __restrict__ z,
                         const float* __restrict__ Win1,
                         float* __restrict__ xnew_out,
                         float* __restrict__ phi_xn,
                         float* __restrict__ psi_z) {
    __shared__ float s_phi[256];
    int tid = threadIdx.x;
    s_phi[tid] = sigmoidf(x[tid]);
    psi_z[tid]       = tanhf(z[tid]);
    psi_z[tid + 256] = tanhf(z[tid + 256]);
    __syncthreads();
    float accw = 0.f;
    const float* wr = W + (size_t)tid * 256;
    #pragma unroll 8
    for (int j = 0; j < 256; ++j) accw += wr[j] * s_phi[j];
    float acci = 0.f;
    const float* wir = Win1 + (size_t)tid * 512;
    #pragma unroll 8
    for (int k = 0; k < 512; ++k) acci += wir[k] * I[k];
    float xn = (1.0f - GAMMA) * x[tid] + GAMMA * accw + acci;
    xnew_out[tid] = xn;
    phi_xn[tid] = sigmoidf(xn);
}

// Kernel 2: W_new = 0.9*W + 0.1*(C*Phi + D@psi_z), one 16-row WMMA tile per wave.
__global__ void k2_wnew(const float* __restrict__ D, const float* __restrict__ W,
                        const float* __restrict__ C,
                        const float* __restrict__ psi_z,
                        const float* __restrict__ phi_xn,
                        float* __restrict__ wnew_out) {
    __shared__ float s_psi[512];
    __shared__ float s_phi[256];
    int tid = threadIdx.x;
    s_psi[tid]       = psi_z[tid];
    s_psi[tid + 256] = psi_z[tid + 256];
    s_phi[tid]       = phi_xn[tid];
    __syncthreads();

    int lane = tid & 31;
    int half = lane >> 4;           // K-pair select (A layout: lanes 16-31 hold K=2,3)
    int m    = lane & 15;           // row within tile
    int row0 = (blockIdx.x * 8 + (tid >> 5)) * 16;

    const float* arow = D + (size_t)(row0 + m) * 512 + 2 * half;
    v8f c = {};
    #pragma unroll 4
    for (int k0 = 0; k0 < 512; k0 += 4) {
        v2f a = *(const v2f*)(arow + k0);             // D[row][k0+2h], D[row][k0+2h+1]
        v2f b;                                        // psi chunk replicated to all cols
        b.x = s_psi[k0 + 2 * half];
        b.y = s_psi[k0 + 2 * half + 1];
        c = __builtin_amdgcn_wmma_f32_16x16x4_f32(false, a, false, b, (short)0, c,
                                                  false, false);
    }
    if (m == 0) {                   // lane 0 -> rows 0..7, lane 16 -> rows 8..15
        #pragma unroll
        for (int j = 0; j < 8; ++j) {
            int row = row0 + half * 8 + j;
            float phi = s_phi[row >> 8] * s_phi[row & 255];
            wnew_out[row] = (1.0f - GAMMA) * W[row] + GAMMA * (C[row] * phi + c[j]);
        }
    }
}

// Kernel 3: partial[ksplit][row] = H[row, ksplit*1024 : +1024] @ Phi-chunk (WMMA).
__global__ void k3_hphi(const float* __restrict__ H, const float* __restrict__ phi_xn,
                        float* __restrict__ partial) {
    __shared__ float s_phi[256];
    int tid = threadIdx.x;
    s_phi[tid] = phi_xn[tid];
    __syncthreads();

    int wave_g = blockIdx.x * 8 + (tid >> 5);
    int tile   = wave_g >> 6;       // 0..31  (row tile)
    int ksplit = wave_g & 63;       // 0..63  (K split)
    int lane = tid & 31;
    int half = lane >> 4;
    int m    = lane & 15;
    int row0 = tile * 16;
    int kstart = ksplit * 1024;

    const float* arow = H + (size_t)(row0 + m) * 65536 + kstart + 2 * half;
    v8f c = {};
    for (int blk = 0; blk < 4; ++blk) {
        int kbase = blk * 256;
        float hi = s_phi[(kstart + kbase) >> 8];      // uniform within 256-block
        __builtin_prefetch(arow + kbase + 256, 0, 1);
        #pragma unroll 4
        for (int k0 = kbase; k0 < kbase + 256; k0 += 4) {
            v2f a = *(const v2f*)(arow + k0);
            int kl = (k0 & 255) + 2 * half;           // even, so kl+1 <= 255
            v2f b;
            b.x = hi * s_phi[kl];
            b.y = hi * s_phi[kl + 1];
            c = __builtin_amdgcn_wmma_f32_16x16x4_f32(false, a, false, b, (short)0, c,
                                                      false, false);
        }
    }
    if (m == 0) {
        #pragma unroll
        for (int j = 0; j < 8; ++j) {
            int row = row0 + half * 8 + j;
            partial[ksplit * 512 + row] = c[j];
        }
    }
}

// Kernel 4: z_new = 0.999*z + 0.001*(F*psi_z + sum(partials) + W_in_2@I)
__global__ void k4_znew(const float* __restrict__ I, const float* __restrict__ z,
                        const float* __restrict__ F, const float* __restrict__ Win2,
                        const float* __restrict__ psi_z,
                        const float* __restrict__ partial,
                        float* __restrict__ znew_out) {
    int mrow = blockIdx.x * blockDim.x + threadIdx.x;  // 0..511
    float hsum = 0.f;
    #pragma unroll 8
    for (int s = 0; s < 64; ++s) hsum += partial[s * 512 + mrow];
    float din = 0.f;
    const float* wr = Win2 + (size_t)mrow * 512;
    #pragma unroll 8
    for (int k = 0; k < 512; ++k) din += wr[k] * I[k];
    const float gt = GAMMA * TAU;
    znew_out[mrow] = (1.0f - gt) * z[mrow] + gt * (F[mrow] * psi_z[mrow] + hsum + din);
}

extern "C" void kernel_launch(void* const* d_in, const int* in_sizes, int n_in,
                              void* d_out, int out_size, void* d_ws, size_t ws_size,
                              hipStream_t stream) {
    const float* I    = (const float*)d_in[0];
    const float* x    = (const float*)d_in[1];
    const float* W    = (const float*)d_in[2];
    const float* z    = (const float*)d_in[3];
    const float* C    = (const float*)d_in[4];
    const float* D    = (const float*)d_in[5];
    const float* F    = (const float*)d_in[6];
    const float* H    = (const float*)d_in[7];
    const float* Win1 = (const float*)d_in[8];
    const float* Win2 = (const float*)d_in[9];
    (void)in_sizes; (void)n_in; (void)out_size; (void)ws_size;

    float* out  = (float*)d_out;
    float* xnew = out;                   // 256
    float* wnew = out + 256;             // 65536
    float* znew = out + 256 + 65536;     // 512

    float* ws      = (float*)d_ws;
    float* phi_xn  = ws;                 // 256
    float* psi_z   = ws + 256;           // 512
    float* partial = ws + 1024;          // 64*512

    k1_state<<<1,   256, 0, stream>>>(I, x, W, z, Win1, xnew, phi_xn, psi_z);
    k2_wnew <<<512, 256, 0, stream>>>(D, W, C, psi_z, phi_xn, wnew);
    k3_hphi <<<256, 256, 0, stream>>>(H, phi_xn, partial);
    k4_znew <<<2,   256, 0, stream>>>(I, z, F, Win2, psi_z, partial, znew);
}